// Assigner_85950885527720
// MI455X (gfx1250) — compile-verified
//
#include <hip/hip_runtime.h>
#include <hip/hip_bf16.h>
#include <math.h>

// ---------------------------------------------------------------------------
// Problem constants (from reference): bs=16, na=8400, n_gt=64, NC=80, TOPK=10
// ---------------------------------------------------------------------------
#define BS    16
#define NA    8400
#define NGT   64
#define NCLS  80
#define TOPKN 10
#define TILE  64             // anchors per block in metric kernel (2 waves)
#define EPSF  1e-9f

// ---------------------------------------------------------------------------
// CDNA5 feature detection (guarded so the file always compiles; #warning
// probes confirm via stderr whether the gfx1250 paths are active)
// ---------------------------------------------------------------------------
#if defined(__has_builtin)
#if __has_builtin(__builtin_amdgcn_tensor_load_to_lds) && \
    __has_builtin(__builtin_amdgcn_s_wait_tensorcnt)
#define USE_TDM 1
#endif
#if __has_builtin(__builtin_amdgcn_wmma_f32_16x16x4_f32)
#define USE_WMMA 1
#endif
#endif
#ifndef USE_TDM
#define USE_TDM 0
#endif
#ifndef USE_WMMA
#define USE_WMMA 0
#endif

#if defined(__AMDGCN__)
#if !USE_TDM
#warning "gfx1250 TDM builtins unavailable: cooperative-load fallback in use"
#endif
#if !USE_WMMA
#warning "gfx1250 WMMA f32 builtin unavailable: scalar LDS gather in use"
#endif
#endif

typedef float v2f __attribute__((ext_vector_type(2)));
typedef float v8f __attribute__((ext_vector_type(8)));

#if USE_TDM
typedef unsigned int v4u __attribute__((ext_vector_type(4)));
typedef int          v4i __attribute__((ext_vector_type(4)));
typedef int          v8i __attribute__((ext_vector_type(8)));

// 1D contiguous tile: tensor_dim0 = valid elems (OOB reads return 0),
// tile_dim0 = tile elems, data_size = 4 bytes, tile_dim1/2 = 0 (unused).
__device__ __forceinline__ void tdm_load_1d(const void* gsrc, unsigned lds_addr,
                                            unsigned tensor_elems,
                                            unsigned tile_elems)
{
  unsigned long long ga = (unsigned long long)(size_t)gsrc;
  v4u g0;
  g0[0] = 1u;                                           // count=1, user desc
  g0[1] = lds_addr;                                     // LDS byte address
  g0[2] = (unsigned)ga;                                 // global_addr[31:0]
  g0[3] = ((unsigned)(ga >> 32) & 0x01ffffffu)          // global_addr[56:32]
          | (2u << 30);                                 // type=2 ("image")
  v8i g1;
  g1[0] = (int)(2u << 16);                              // data_size=2 -> 4B
  g1[1] = (int)((tensor_elems & 0xffffu) << 16);        // tensor_dim0[15:0]
  g1[2] = (int)(((tensor_elems >> 16) & 0xffffu)        // tensor_dim0[31:16]
                | (1u << 16));                          // tensor_dim1 = 1
  g1[3] = (int)((tile_elems & 0xffffu) << 16);          // tile_dim0
  g1[4] = 0;                                            // tile_dim1/2 unused
  g1[5] = (int)tensor_elems;                            // dim0 stride (unused)
  g1[6] = 0;
  g1[7] = 0;
  v4i z4 = {0, 0, 0, 0};
#if __has_include(<hip/amd_detail/amd_gfx1250_TDM.h>)
  v8i z8 = {0, 0, 0, 0, 0, 0, 0, 0};
  __builtin_amdgcn_tensor_load_to_lds(g0, g1, z4, z4, z8, 0);
#else
  __builtin_amdgcn_tensor_load_to_lds(g0, g1, z4, z4, 0);
#endif
}
#endif // USE_TDM

// ---------------------------------------------------------------------------
// CIoU, exactly mirroring the reference (box1 = gt, box2 = pred)
// ---------------------------------------------------------------------------
__device__ __forceinline__ float ciou_f(float gx1, float gy1, float gx2, float gy2,
                                        float px1, float py1, float px2, float py2)
{
  const float eps = 1e-7f;
  float w1 = gx2 - gx1, h1 = gy2 - gy1 + eps;
  float w2 = px2 - px1, h2 = py2 - py1 + eps;
  float iw = fmaxf(fminf(gx2, px2) - fmaxf(gx1, px1), 0.f);
  float ih = fmaxf(fminf(gy2, py2) - fmaxf(gy1, py1), 0.f);
  float inter = iw * ih;
  float uni = w1 * h1 + w2 * h2 - inter + eps;
  float iou = inter / uni;
  float cw = fmaxf(gx2, px2) - fminf(gx1, px1);
  float ch = fmaxf(gy2, py2) - fminf(gy1, py1);
  float c2 = cw * cw + ch * ch + eps;
  float dx = px1 + px2 - gx1 - gx2;
  float dy = py1 + py2 - gy1 - gy2;
  float rho2 = (dx * dx + dy * dy) * 0.25f;
  float t = atanf(w2 / h2) - atanf(w1 / h1);
  float v = 0.40528473456935109f * t * t;               // 4/pi^2
  float alpha = v / (v - iou + 1.0f + eps);
  return iou - (rho2 / c2 + v * alpha);
}

// overlaps = where(in_gts && valid, clip(ciou, 0), 0)
__device__ __forceinline__ float masked_ovl(const float* __restrict__ gt_box,
                                            const float* __restrict__ gmask,
                                            int b, int n, float ax, float ay,
                                            float4 pb)
{
  if (gmask[b * NGT + n] <= 0.f) return 0.f;
  const float* g = gt_box + ((size_t)b * NGT + n) * 4;
  float gx1 = g[0], gy1 = g[1], gx2 = g[2], gy2 = g[3];
  float din = fminf(fminf(ax - gx1, ay - gy1), fminf(gx2 - ax, gy2 - ay));
  if (din <= EPSF) return 0.f;
  return fmaxf(ciou_f(gx1, gy1, gx2, gy2, pb.x, pb.y, pb.z, pb.w), 0.f);
}

// ---------------------------------------------------------------------------
// Kernel 0: zero assignment bitmasks and per-GT max accumulators
// ---------------------------------------------------------------------------
__global__ __launch_bounds__(256) void k_init(unsigned long long* __restrict__ posBits,
                                              unsigned* __restrict__ posMet,
                                              unsigned* __restrict__ posOvl)
{
  int i = blockIdx.x * 256 + threadIdx.x;
  if (i < BS * NA) posBits[i] = 0ull;
  if (i < BS * NGT) { posMet[i] = 0u; posOvl[i] = 0u; }
}

// ---------------------------------------------------------------------------
// Kernel 1: metric[b][n][a].
//  - TDM async-stages the 64x80 f32 score tile into LDS (TENSORcnt).
//  - The label gather is computed as OneHot(64x80) x Score^T(80x64) on the
//    matrix pipe with V_WMMA_F32_16X16X4_F32 (bit-exact: one-hot rows mean
//    each output element accumulates exactly one unrounded fp32 product).
// ---------------------------------------------------------------------------
__global__ __launch_bounds__(TILE) void k_metric(const float* __restrict__ score,
                                                 const float* __restrict__ p_box,
                                                 const float* __restrict__ anchors,
                                                 const int*   __restrict__ gt_labels,
                                                 const float* __restrict__ gt_box,
                                                 const float* __restrict__ gmask,
                                                 float* __restrict__ metric)
{
  __shared__ float s_score[TILE * NCLS];   // 20 KB score tile (anchor-major)
  __shared__ float s_gath[NGT * TILE];     // 16 KB gathered[n][a_local]
  __shared__ float s_gt[NGT * 4];
  __shared__ int   s_lab[NGT];
  __shared__ float s_msk[NGT];

  const int b  = blockIdx.y;
  const int a0 = blockIdx.x * TILE;
  const int t  = threadIdx.x;              // 0..63 (2 waves)

#if USE_TDM
  if (t < 32) {  // wave 0 issues the async tensor DMA
    const float* src = score + ((size_t)b * NA + a0) * NCLS;
    unsigned remain = (unsigned)((NA - a0) * NCLS);
    if (remain > TILE * NCLS) remain = TILE * NCLS;
    tdm_load_1d((const void*)src, (unsigned)(size_t)&s_score[0],
                remain, TILE * NCLS);
    __builtin_amdgcn_s_wait_tensorcnt(0);
  }
#else
  for (int i = t; i < TILE * NCLS; i += TILE) {
    int aa = a0 + i / NCLS;
    s_score[i] = (aa < NA) ? score[((size_t)b * NA + a0) * NCLS + i] : 0.f;
  }
#endif

  {  // stage GT data: t covers 0..63 exactly
    const float* g = gt_box + ((size_t)b * NGT + t) * 4;
    s_gt[t * 4 + 0] = g[0]; s_gt[t * 4 + 1] = g[1];
    s_gt[t * 4 + 2] = g[2]; s_gt[t * 4 + 3] = g[3];
    s_lab[t] = gt_labels[b * NGT + t];
    s_msk[t] = gmask[b * NGT + t];
  }
  __syncthreads();

#if USE_WMMA
  {
    const int w    = t >> 5;    // wave id (0..1), uniform per wave
    const int l    = t & 31;    // lane id (wave32)
    const int nCol = l & 15;    // N index within 16x16 tile
    const int h    = l >> 4;    // lane-half
    for (int mt = 0; mt < 2; ++mt) {
      const int m0   = (w * 2 + mt) * 16;       // GT-row tile base
      const int labL = s_lab[m0 + (l & 15)];    // A row M = l&15 (both halves)
      for (int nt = 0; nt < 4; ++nt) {
        const int i0 = nt * 16;                 // anchor-col tile base
        v8f acc = {0.f, 0.f, 0.f, 0.f, 0.f, 0.f, 0.f, 0.f};
        for (int ks = 0; ks < 20; ++ks) {       // K = 80 in steps of 4
          const int k0 = ks * 4;
          // A (16x4): lane l holds M=l&15, K = 2*h + v  (v = vector elem)
          v2f av;
          av.x = (labL == (k0 + 2 * h + 0)) ? 1.f : 0.f;
          av.y = (labL == (k0 + 2 * h + 1)) ? 1.f : 0.f;
          // B (4x16): lane l holds N=l&15, K = v + 2*h -> contiguous float2
          const v2f bv =
              *(const v2f*)&s_score[(i0 + nCol) * NCLS + k0 + 2 * h];
          acc = __builtin_amdgcn_wmma_f32_16x16x4_f32(
              false, av, false, bv, (short)0, acc, false, false);
        }
        // D (16x16): VGPR r, lane l -> M = r + 8*h, N = l&15
#pragma unroll
        for (int r = 0; r < 8; ++r)
          s_gath[(m0 + r + 8 * h) * TILE + i0 + nCol] = acc[r];
      }
    }
  }
  __syncthreads();
#endif

  const int a = a0 + t;
  float4 pb = make_float4(0.f, 0.f, 0.f, 0.f);
  float ax = 0.f, ay = 0.f;
  if (a < NA) {
    pb = ((const float4*)p_box)[(size_t)b * NA + a];
    ax = anchors[a * 2 + 0];
    ay = anchors[a * 2 + 1];
  }

  for (int n = 0; n < NGT; ++n) {
    float gx1 = s_gt[n * 4 + 0], gy1 = s_gt[n * 4 + 1];
    float gx2 = s_gt[n * 4 + 2], gy2 = s_gt[n * 4 + 3];
    float met = 0.f;
    if (a < NA && s_msk[n] > 0.f) {
      float din = fminf(fminf(ax - gx1, ay - gy1), fminf(gx2 - ax, gy2 - ay));
      if (din > EPSF) {
        float c = ciou_f(gx1, gy1, gx2, gy2, pb.x, pb.y, pb.z, pb.w);
        float o = fmaxf(c, 0.f);
#if USE_WMMA
        float sc = s_gath[n * TILE + t];       // conflict-free broadcast rows
#else
        float sc = s_score[t * NCLS + s_lab[n]];
#endif
        met = sqrtf(sc) * powf(o, 6.0f);       // score^ALPHA * ovl^BETA
      }
    }
    if (a < NA) metric[((size_t)b * NGT + n) * NA + a] = met;
  }
}

// ---------------------------------------------------------------------------
// Kernel 2: exact top-10 per valid (b,n) row; scatter assignments as bits.
// Tie-break matches jax.lax.top_k: higher value first, lower index on ties.
// ---------------------------------------------------------------------------
__global__ __launch_bounds__(256) void k_topk(const float* __restrict__ metric,
                                              const float* __restrict__ anchors,
                                              const float* __restrict__ gt_box,
                                              const float* __restrict__ gmask,
                                              unsigned long long* __restrict__ posBits)
{
  const int bn = blockIdx.x;
  const int b  = bn / NGT;
  const int n  = bn - b * NGT;
  if (gmask[bn] <= 0.f) return;                  // uniform: invalid GT row

  const float* row = metric + (size_t)bn * NA;
  __shared__ unsigned long long s_red[256];
  __shared__ int s_sel[TOPKN];
  const int t = threadIdx.x;

  for (int k = 0; k < TOPKN; ++k) {
    unsigned long long best = 0ull;
    for (int a = t; a < NA; a += 256) {
      bool excl = false;
      for (int j = 0; j < k; ++j) excl |= (s_sel[j] == a);
      if (!excl) {
        // metric >= 0 -> float bits are order-preserving as uint
        unsigned long long key =
            ((unsigned long long)__float_as_uint(row[a]) << 32) |
            (unsigned)(NA - 1 - a);
        if (key > best) best = key;
      }
    }
    s_red[t] = best;
    __syncthreads();
    for (int s = 128; s > 0; s >>= 1) {
      if (t < s) {
        unsigned long long o = s_red[t + s];
        if (o > s_red[t]) s_red[t] = o;
      }
      __syncthreads();
    }
    if (t == 0) s_sel[k] = NA - 1 - (int)(s_red[0] & 0xffffffffu);
    __syncthreads();
  }

  if (t < TOPKN) {
    int a = s_sel[t];
    float ax = anchors[a * 2 + 0], ay = anchors[a * 2 + 1];
    const float* g = gt_box + (size_t)bn * 4;
    float din = fminf(fminf(ax - g[0], ay - g[1]), fminf(g[2] - ax, g[3] - ay));
    if (din > EPSF)
      atomicOr(&posBits[(size_t)b * NA + a], 1ull << n);
  }
}

// ---------------------------------------------------------------------------
// Kernel 3: per-anchor conflict resolution + per-GT max accumulators
// ---------------------------------------------------------------------------
__global__ __launch_bounds__(256) void k_resolve(const unsigned long long* __restrict__ posBits,
                                                 const float* __restrict__ metric,
                                                 const float* __restrict__ p_box,
                                                 const float* __restrict__ anchors,
                                                 const float* __restrict__ gt_box,
                                                 const float* __restrict__ gmask,
                                                 int* __restrict__ gtIdx,
                                                 int* __restrict__ fgArr,
                                                 unsigned* __restrict__ posMet,
                                                 unsigned* __restrict__ posOvl)
{
  int idx = blockIdx.x * 256 + threadIdx.x;
  if (idx >= BS * NA) return;
  int b = idx / NA, a = idx - b * NA;

  unsigned long long bits = posBits[idx];
  int fg = __popcll(bits);
  int g = 0;
  float ax = anchors[a * 2 + 0], ay = anchors[a * 2 + 1];
  float4 pb = ((const float4*)p_box)[idx];

  if (fg > 1) {
    // mask_pos := one_hot(argmax_n overlaps[b,:,a]) — first max wins
    float best = -1.f;
    for (int nn = 0; nn < NGT; ++nn) {
      float o = masked_ovl(gt_box, gmask, b, nn, ax, ay, pb);
      if (o > best) { best = o; g = nn; }
    }
    fg = 1;
  } else if (fg == 1) {
    g = (int)__builtin_ctzll(bits);
  }

  gtIdx[idx] = g;
  fgArr[idx] = fg;

  if (fg) {
    float m = metric[((size_t)b * NGT + g) * NA + a];
    float o = masked_ovl(gt_box, gmask, b, g, ax, ay, pb);
    // values >= 0: uint compare == float compare; deterministic max
    atomicMax(&posMet[b * NGT + g], __float_as_uint(m));
    atomicMax(&posOvl[b * NGT + g], __float_as_uint(o));
  }
}

// ---------------------------------------------------------------------------
// Kernel 4: final outputs (bboxes for ALL anchors via gt_idx, one-hot scores,
// fg mask)
// ---------------------------------------------------------------------------
__global__ __launch_bounds__(256) void k_final(const int* __restrict__ gtIdx,
                                               const int* __restrict__ fgArr,
                                               const float* __restrict__ metric,
                                               const unsigned* __restrict__ posMet,
                                               const unsigned* __restrict__ posOvl,
                                               const float* __restrict__ gt_box,
                                               const int* __restrict__ gt_labels,
                                               float* __restrict__ out_bbox,
                                               float* __restrict__ out_scores,
                                               float* __restrict__ out_fg)
{
  int idx = blockIdx.x * 256 + threadIdx.x;
  if (idx >= BS * NA) return;
  int b = idx / NA, a = idx - b * NA;
  int g  = gtIdx[idx];
  int fg = fgArr[idx];

  const float* gb = gt_box + ((size_t)b * NGT + g) * 4;
  float4 bb = make_float4(gb[0], gb[1], gb[2], gb[3]);
  ((float4*)out_bbox)[idx] = bb;                  // written even when fg==0
  out_fg[idx] = fg ? 1.f : 0.f;

  float norm = 0.f;
  int lab = 0;
  if (fg) {
    lab = gt_labels[b * NGT + g];
    if (lab < 0) lab = 0;
    float m  = metric[((size_t)b * NGT + g) * NA + a];
    float pm = __uint_as_float(posMet[b * NGT + g]);
    float po = __uint_as_float(posOvl[b * NGT + g]);
    norm = m * po / (pm + EPSF);
  }

  float4* sc = (float4*)(out_scores + (size_t)idx * NCLS);
#pragma unroll
  for (int c4 = 0; c4 < NCLS / 4; ++c4) {
    float4 v = make_float4(0.f, 0.f, 0.f, 0.f);
    if (fg && (lab >> 2) == c4) ((float*)&v)[lab & 3] = norm;
    sc[c4] = v;
  }
}

// ---------------------------------------------------------------------------
// Host launcher
// ---------------------------------------------------------------------------
extern "C" void kernel_launch(void* const* d_in, const int* in_sizes, int n_in,
                              void* d_out, int out_size, void* d_ws, size_t ws_size,
                              hipStream_t stream)
{
  (void)in_sizes; (void)n_in; (void)out_size; (void)ws_size;

  const float* score     = (const float*)d_in[0];  // (16,8400,80)
  const float* p_box     = (const float*)d_in[1];  // (16,8400,4)
  const float* anchors   = (const float*)d_in[2];  // (8400,2)
  const int*   gt_labels = (const int*)  d_in[3];  // (16,64,1)
  const float* gt_box    = (const float*)d_in[4];  // (16,64,4)
  const float* gmask     = (const float*)d_in[5];  // (16,64,1)

  float* out        = (float*)d_out;
  float* out_bbox   = out;                                   // 16*8400*4
  float* out_scores = out + (size_t)BS * NA * 4;             // 16*8400*80
  float* out_fg     = out_scores + (size_t)BS * NA * NCLS;   // 16*8400

  // Workspace layout (~36.6 MB)
  float* metric = (float*)d_ws;                                       // BS*NGT*NA f32
  unsigned long long* posBits =
      (unsigned long long*)(metric + (size_t)BS * NGT * NA);          // BS*NA u64
  unsigned* posMet = (unsigned*)(posBits + (size_t)BS * NA);          // BS*NGT u32
  unsigned* posOvl = posMet + BS * NGT;                               // BS*NGT u32
  int* gtIdx = (int*)(posOvl + BS * NGT);                             // BS*NA i32
  int* fgArr = gtIdx + (size_t)BS * NA;                               // BS*NA i32

  const int anchorBlocks = (BS * NA + 255) / 256;                     // 525 exact

  k_init<<<anchorBlocks, 256, 0, stream>>>(posBits, posMet, posOvl);

  dim3 gridM((NA + TILE - 1) / TILE, BS);                             // 132 x 16
  k_metric<<<gridM, TILE, 0, stream>>>(score, p_box, anchors, gt_labels,
                                       gt_box, gmask, metric);

  k_topk<<<BS * NGT, 256, 0, stream>>>(metric, anchors, gt_box, gmask, posBits);

  k_resolve<<<anchorBlocks, 256, 0, stream>>>(posBits, metric, p_box, anchors,
                                              gt_box, gmask, gtIdx, fgArr,
                                              posMet, posOvl);

  k_final<<<anchorBlocks, 256, 0, stream>>>(gtIdx, fgArr, metric, posMet, posOvl,
                                            gt_box, gt_labels,
                                            out_bbox, out_scores, out_fg);
}